// CausalConv1D_74337293959631
// MI455X (gfx1250) — compile-verified
//
#include <hip/hip_runtime.h>
#include <cstdint>
#include <cstddef>

// Causal depthwise conv1d, TW=4 taps, fp32, bandwidth-bound.
// Tile: 16 timesteps x 512 channels per block, staged in LDS via the
// CDNA5 Tensor Data Mover (tensor_load_to_lds), computed with a rolling
// 4-deep register window, written back with non-temporal b128 stores.

#define TW 4                       // TEMPORAL_WIDTH
#define TT 16                      // timesteps per block tile
#define CC 512                     // channels per block tile
#define CCV (CC / 4)               // float4 per LDS row = 128
#define ROWS (TT + TW - 1)         // 19 staged time rows
#define NTHREADS 256

typedef __attribute__((ext_vector_type(4))) float        f32x4;
typedef __attribute__((ext_vector_type(4))) unsigned int u32x4;
typedef __attribute__((ext_vector_type(8))) int          i32x8;
typedef __attribute__((ext_vector_type(4))) int          i32x4;

__global__ __launch_bounds__(NTHREADS) void causal_conv_kernel(
    const float* __restrict__ x, const float* __restrict__ cache,
    const float* __restrict__ w, const float* __restrict__ bias,
    float* __restrict__ y, int B, int T, int W)
{
    __shared__ f32x4 tile[ROWS * CCV];   // 19 x 512 floats = 38,912 B

    const int nChunk = W / CC;
    const int nTiles = T / TT;
    int bid = blockIdx.x;
    const int chunk = bid % nChunk; bid /= nChunk;
    const int tileI = bid % nTiles;
    const int b     = bid / nTiles;
    const int t0 = tileI * TT;
    const int c0 = chunk * CC;

    // LDS row r holds x_cat time (t0 + r). For t0>0 all 19 rows come from x
    // starting at t0-3; for t0==0 rows 0..2 come from cache, rows 3..18 from x.
    const int tstart  = (t0 == 0) ? 0 : (t0 - (TW - 1));
    const int nrows   = (t0 == 0) ? TT : ROWS;
    const int ldsRow0 = (t0 == 0) ? (TW - 1) : 0;

    const float* xblk = x + ((size_t)b * T + (size_t)tstart) * W + c0;

#if defined(__AMDGCN__) && __has_builtin(__builtin_amdgcn_tensor_load_to_lds) && __has_builtin(__builtin_amdgcn_s_wait_tensorcnt)
    if (threadIdx.x < 32u) {
        // One wave issues a single TDM descriptor: 2D tile, nrows x CC fp32
        // elements, global row stride = W elements, packed into LDS.
        uint64_t ga  = (uint64_t)(uintptr_t)xblk;
        uint32_t lds = (uint32_t)(uintptr_t)(&tile[0])
                     + (uint32_t)(ldsRow0 * CCV) * 16u;
        // D# group 0: count=1 | lds_addr | global_addr[56:0] | type=2
        u32x4 g0 = { 1u,
                     lds,
                     (uint32_t)ga,
                     (uint32_t)((ga >> 32) & 0x01FFFFFFull) | 0x80000000u };
        // D# group 1: data_size=4B; tensor_dim0=CC; tensor_dim1=nrows;
        // tile_dim0=CC; tile_dim1=nrows; tensor_dim0_stride=W
        i32x8 g1 = { (int)0x00020000u,
                     (int)((uint32_t)CC    << 16),
                     (int)((uint32_t)nrows << 16),
                     (int)((uint32_t)CC    << 16),
                     nrows,
                     W,
                     0, 0 };
        i32x4 gz4 = { 0, 0, 0, 0 };
        i32x8 gz8 = { 0, 0, 0, 0, 0, 0, 0, 0 };
        __builtin_amdgcn_tensor_load_to_lds(g0, g1, gz4, gz4, gz8, 0);
        __builtin_amdgcn_s_wait_tensorcnt(0);
    }
    if (t0 == 0) {   // stage 3 cache rows while the TDM fills x rows
        const f32x4* cb = (const f32x4*)(cache + ((size_t)b * (TW - 1)) * W + c0);
        for (int idx = (int)threadIdx.x; idx < (TW - 1) * CCV; idx += NTHREADS) {
            int r = idx / CCV, cv = idx % CCV;
            tile[r * CCV + cv] = cb[(size_t)r * (W / 4) + cv];
        }
    }
#else
    {   // fallback: cooperative global->LDS staging
        const f32x4* xb4 = (const f32x4*)xblk;
        for (int idx = (int)threadIdx.x; idx < nrows * CCV; idx += NTHREADS) {
            int r = idx / CCV, cv = idx % CCV;
            tile[(ldsRow0 + r) * CCV + cv] = xb4[(size_t)r * (W / 4) + cv];
        }
        if (t0 == 0) {
            const f32x4* cb = (const f32x4*)(cache + ((size_t)b * (TW - 1)) * W + c0);
            for (int idx = (int)threadIdx.x; idx < (TW - 1) * CCV; idx += NTHREADS) {
                int r = idx / CCV, cv = idx % CCV;
                tile[r * CCV + cv] = cb[(size_t)r * (W / 4) + cv];
            }
        }
    }
#endif
    __syncthreads();

    // 256 threads: cv = float4 channel column (0..127), th picks 8 timesteps.
    const int cv    = (int)threadIdx.x & (CCV - 1);
    const int th    = (int)threadIdx.x >> 7;
    const int tbase = th * (TT / 2);
    const int c4    = c0 + cv * 4;

    const f32x4 wv0 = *(const f32x4*)(w + 0 * (size_t)W + c4);
    const f32x4 wv1 = *(const f32x4*)(w + 1 * (size_t)W + c4);
    const f32x4 wv2 = *(const f32x4*)(w + 2 * (size_t)W + c4);
    const f32x4 wv3 = *(const f32x4*)(w + 3 * (size_t)W + c4);
    const f32x4 bv  = *(const f32x4*)(bias + c4);

    // Rolling window: y[t0+tt] = bv + tile[tt]*w0 + tile[tt+1]*w1 + ...
    f32x4 a0 = tile[(tbase + 0) * CCV + cv];
    f32x4 a1 = tile[(tbase + 1) * CCV + cv];
    f32x4 a2 = tile[(tbase + 2) * CCV + cv];

    #pragma unroll
    for (int i = 0; i < TT / 2; ++i) {
        f32x4 a3 = tile[(tbase + 3 + i) * CCV + cv];
        f32x4 r = bv;
        r += a0 * wv0;
        r += a1 * wv1;
        r += a2 * wv2;
        r += a3 * wv3;
        float* yp = y + ((size_t)b * T + (size_t)(t0 + tbase + i)) * W + c4;
        __builtin_nontemporal_store(r, (f32x4*)yp);   // y is write-once: keep it out of L2
        a0 = a1; a1 = a2; a2 = a3;
    }
}

// new_cache = x[:, T-3:, :]
__global__ __launch_bounds__(NTHREADS) void cache_tail_kernel(
    const float* __restrict__ x, float* __restrict__ outc, int B, int T, int W)
{
    const int W4  = W / 4;
    const int per = (TW - 1) * W4;
    const int n   = B * per;
    int i = (int)(blockIdx.x * NTHREADS + threadIdx.x);
    if (i >= n) return;
    int b = i / per, rem = i % per;
    int r = rem / W4, cvv = rem % W4;
    const f32x4* xs = (const f32x4*)x;
    ((f32x4*)outc)[i] = xs[((size_t)b * T + (size_t)(T - (TW - 1) + r)) * W4 + cvv];
}

extern "C" void kernel_launch(void* const* d_in, const int* in_sizes, int n_in,
                              void* d_out, int out_size, void* d_ws, size_t ws_size,
                              hipStream_t stream) {
    const float* x     = (const float*)d_in[0];
    const float* cache = (const float*)d_in[1];
    const float* w     = (const float*)d_in[2];
    const float* bias  = (const float*)d_in[3];
    float* y = (float*)d_out;

    const int W = in_sizes[3];                       // bias has W elements
    const int B = in_sizes[1] / ((TW - 1) * W);      // cache: B*(TW-1)*W
    const int T = in_sizes[0] / (B * W);             // x: B*T*W
    float* outc = y + (size_t)B * T * W;             // outputs concatenated: y, new_cache

    const int nChunk = W / CC;
    const int nTiles = T / TT;
    causal_conv_kernel<<<dim3((unsigned)(B * nTiles * nChunk)), dim3(NTHREADS), 0, stream>>>(
        x, cache, w, bias, y, B, T, W);

    const int n4 = B * (TW - 1) * (W / 4);
    cache_tail_kernel<<<dim3((unsigned)((n4 + NTHREADS - 1) / NTHREADS)), dim3(NTHREADS), 0, stream>>>(
        x, outc, B, T, W);
}